// NetVLAD_5961414606919
// MI455X (gfx1250) — compile-verified
//
#include <hip/hip_runtime.h>
#include <hip/hip_bf16.h>

#define BB 32
#define NN 4096
#define DD 256
#define KK 128
#define DC 64

typedef float v2f __attribute__((ext_vector_type(2)));
typedef float v4f __attribute__((ext_vector_type(4)));
typedef float v8f __attribute__((ext_vector_type(8)));

__device__ __forceinline__ v8f wmma_f32(v2f a, v2f b, v8f c) {
  // V_WMMA_F32_16X16X4_F32 : D = A(16x4) * B(4x16) + C(16x16), all f32
  return __builtin_amdgcn_wmma_f32_16x16x4_f32(false, a, false, b, (short)0, c,
                                               false, false);
}

// ---------------------------------------------------------------------------
// Kernel 0: zero the norm accumulator
// ---------------------------------------------------------------------------
__global__ void nv_init_kernel(float* __restrict__ normAcc) {
  if (threadIdx.x < BB) normAcc[threadIdx.x] = 0.0f;
}

// ---------------------------------------------------------------------------
// Kernel 1: transpose weights so WMMA B-tile loads are coalesced
//   W1 (64,256) -> W1T (256,64) ; Wa (128,256) -> WaT (256,128)
//   W2 (64,64)  -> W2T (64,64)
// ---------------------------------------------------------------------------
__global__ void nv_transpose_kernel(const float* __restrict__ W1,
                                    const float* __restrict__ Wa,
                                    const float* __restrict__ W2,
                                    float* __restrict__ W1T,
                                    float* __restrict__ WaT,
                                    float* __restrict__ W2T) {
  int tid = blockIdx.x * 256 + threadIdx.x;
  if (tid < DC * DD) {  // 16384
    int e = tid / DD, d = tid % DD;
    W1T[d * DC + e] = W1[tid];
  }
  int t2 = tid - DC * DD;
  if (t2 >= 0 && t2 < KK * DD) {  // 32768
    int k = t2 / DD, d = t2 % DD;
    WaT[d * KK + k] = Wa[t2];
  }
  int t3 = t2 - KK * DD;
  if (t3 >= 0 && t3 < DC * DC) {  // 4096
    int f = t3 / DC, e = t3 % DC;
    W2T[e * DC + f] = W2[t3];
  }
}

// ---------------------------------------------------------------------------
// Kernel 2: fused per-token pipeline.
//   one wave (32 lanes) <-> one tile of 16 tokens
//   desc A-tile loaded ONCE per k-step and shared by 12 WMMA accumulators
//   (4 h-tiles for squeeze, 8 logit-tiles for assignment).
//   h -> LDS -> A-layout -> 64x64 squeeze GEMM (WMMA).
//   logits -> LDS -> rowwise softmax (shfl_xor half-wave combine) ->
//   assignT (B,K,N) transposed store for stage 2.
// ---------------------------------------------------------------------------
#define HP 65    // LDS pitch for h  (16 x 64)
#define LP 129   // LDS pitch for logits (16 x 128)

__global__ __launch_bounds__(256) void nv_token_kernel(
    const float* __restrict__ desc, const float* __restrict__ W1T,
    const float* __restrict__ b1, const float* __restrict__ W2T,
    const float* __restrict__ b2, const float* __restrict__ WaT,
    const float* __restrict__ ba, float* __restrict__ sqOut,
    float* __restrict__ assignT) {
  __shared__ float ldsH[8 * 16 * HP];  // 33,280 B
  __shared__ float ldsL[8 * 16 * LP];  // 66,048 B   (total < 100 KB)

  const int tid = threadIdx.x;
  const int wave = tid >> 5;
  const int lane = tid & 31;
  const int r = lane & 15;   // row (token) for A, col for B/C
  const int hf = lane >> 4;  // half-wave selector (K-pair for A/B)

  const int tBase = blockIdx.x * 128 + wave * 16;  // first global token of tile
  const int b = tBase >> 12;                       // / 4096
  const int n0 = tBase & (NN - 1);

  const float* descRow = desc + (size_t)(tBase + r) * DD;

  v8f hAcc[4], lAcc[8];
  const v8f vz = {0.f, 0.f, 0.f, 0.f, 0.f, 0.f, 0.f, 0.f};
#pragma unroll
  for (int t = 0; t < 4; ++t) hAcc[t] = vz;
#pragma unroll
  for (int t = 0; t < 8; ++t) lAcc[t] = vz;

  // ---- phase 1: K-loop over D, desc read once, 12 WMMA per step ----
  for (int d0 = 0; d0 < DD; d0 += 4) {
    v2f a;
    a.x = descRow[d0 + 2 * hf];
    a.y = descRow[d0 + 2 * hf + 1];
#pragma unroll
    for (int t = 0; t < 4; ++t) {
      const float* p = W1T + (size_t)(d0 + hf) * DC + t * 16 + r;
      v2f bb;
      bb.x = p[0];
      bb.y = p[2 * DC];
      hAcc[t] = wmma_f32(a, bb, hAcc[t]);
    }
#pragma unroll
    for (int t = 0; t < 8; ++t) {
      const float* p = WaT + (size_t)(d0 + hf) * KK + t * 16 + r;
      v2f bb;
      bb.x = p[0];
      bb.y = p[2 * KK];
      lAcc[t] = wmma_f32(a, bb, lAcc[t]);
    }
  }

  // ---- bias + LeakyReLU, stage h through LDS (C-layout -> A-layout) ----
  float* myH = ldsH + wave * 16 * HP;
#pragma unroll
  for (int t = 0; t < 4; ++t) {
    float bias = b1[t * 16 + r];
#pragma unroll
    for (int v = 0; v < 8; ++v) {
      float x = hAcc[t][v] + bias;
      x = (x >= 0.f) ? x : 0.01f * x;
      myH[(v + 8 * hf) * HP + t * 16 + r] = x;
    }
  }
  asm volatile("s_wait_dscnt 0x0" ::: "memory");

  // ---- squeeze GEMM: sq = h * W2^T  (16x64 * 64x64) ----
  v8f sAcc[4];
#pragma unroll
  for (int t = 0; t < 4; ++t) sAcc[t] = vz;
  for (int e0 = 0; e0 < DC; e0 += 4) {
    v2f a;
    a.x = myH[r * HP + e0 + 2 * hf];
    a.y = myH[r * HP + e0 + 2 * hf + 1];
#pragma unroll
    for (int t = 0; t < 4; ++t) {
      const float* p = W2T + (size_t)(e0 + hf) * DC + t * 16 + r;
      v2f bb;
      bb.x = p[0];
      bb.y = p[2 * DC];
      sAcc[t] = wmma_f32(a, bb, sAcc[t]);
    }
  }
#pragma unroll
  for (int t = 0; t < 4; ++t) {
    float bias = b2[t * 16 + r];
#pragma unroll
    for (int v = 0; v < 8; ++v) {
      int token = v + 8 * hf;
      sqOut[(size_t)(tBase + token) * DC + t * 16 + r] = sAcc[t][v] + bias;
    }
  }

  // ---- softmax over K=128 per token ----
  float* myL = ldsL + wave * 16 * LP;
#pragma unroll
  for (int t = 0; t < 8; ++t) {
    float bias = ba[t * 16 + r];
#pragma unroll
    for (int v = 0; v < 8; ++v)
      myL[(v + 8 * hf) * LP + t * 16 + r] = lAcc[t][v] + bias;
  }
  asm volatile("s_wait_dscnt 0x0" ::: "memory");

  // lane handles row r, columns [hf*64, hf*64+64)
  float mx = -3.0e38f;
  for (int k = 0; k < 64; ++k) mx = fmaxf(mx, myL[r * LP + hf * 64 + k]);
  mx = fmaxf(mx, __shfl_xor(mx, 16, 32));
  float sum = 0.f;
  for (int k = 0; k < 64; ++k) {
    float e = __expf(myL[r * LP + hf * 64 + k] - mx);
    myL[r * LP + hf * 64 + k] = e;
    sum += e;
  }
  sum += __shfl_xor(sum, 16, 32);
  float inv = 1.f / sum;
  asm volatile("s_wait_dscnt 0x0" ::: "memory");

  // ---- transposed store: assignT[b, k, n0 + r] ----
  float* aT = assignT + (size_t)b * KK * NN + n0 + r;
  for (int k = 0; k < 64; ++k)
    aT[(size_t)(hf * 64 + k) * NN] = myL[r * LP + hf * 64 + k] * inv;
}

// ---------------------------------------------------------------------------
// Kernel 3: per-(batch, ktile) aggregation GEMM
//   agg[k,dc] = sum_n assignT[k,n] * sq[n,dc]   (K-loop = N, via WMMA)
//   mass[k]   = sum_n assignT[k,n]              (free from the A tiles)
//   vlad = agg - mass*centroid ; accumulate squared norm per batch.
// ---------------------------------------------------------------------------
__global__ __launch_bounds__(256) void nv_agg_kernel(
    const float* __restrict__ assignT, const float* __restrict__ sq,
    const float* __restrict__ centroid, float* __restrict__ out,
    float* __restrict__ normAcc) {
  __shared__ float ldsC[8 * 1024];  // 8 waves x (16k x 64dc)
  __shared__ float ldsM[8 * 16];

  const int tid = threadIdx.x;
  const int wave = tid >> 5;
  const int lane = tid & 31;
  const int r = lane & 15;
  const int hf = lane >> 4;
  const int kt = blockIdx.x;  // 0..7
  const int b = blockIdx.y;   // 0..31
  const int k0 = kt * 16;

  const float* aRow = assignT + ((size_t)b * KK + k0 + r) * NN;
  const float* sqB = sq + (size_t)b * NN * DC;

  v8f acc[4];
  const v8f vz = {0.f, 0.f, 0.f, 0.f, 0.f, 0.f, 0.f, 0.f};
#pragma unroll
  for (int t = 0; t < 4; ++t) acc[t] = vz;
  float massP = 0.f;

  const int nBeg = wave * (NN / 8);
  for (int n = nBeg; n < nBeg + NN / 8; n += 4) {
    v2f a;
    a.x = aRow[n + 2 * hf];
    a.y = aRow[n + 2 * hf + 1];
    massP += a.x + a.y;
#pragma unroll
    for (int t = 0; t < 4; ++t) {
      const float* p = sqB + (size_t)(n + hf) * DC + t * 16 + r;
      v2f bb;
      bb.x = p[0];
      bb.y = p[2 * DC];
      acc[t] = wmma_f32(a, bb, acc[t]);
    }
  }

  float* myC = ldsC + wave * 1024;
#pragma unroll
  for (int t = 0; t < 4; ++t)
#pragma unroll
    for (int v = 0; v < 8; ++v)
      myC[(v + 8 * hf) * DC + t * 16 + r] = acc[t][v];
  massP += __shfl_xor(massP, 16, 32);
  if (lane < 16) ldsM[wave * 16 + r] = massP;
  __syncthreads();

  // cross-wave reduce: each thread owns 4 consecutive elements
  const int idx = tid * 4;  // 0..1023
  v4f s = {0.f, 0.f, 0.f, 0.f};
#pragma unroll
  for (int w = 0; w < 8; ++w) {
    const float* p = ldsC + w * 1024 + idx;
    s.x += p[0];
    s.y += p[1];
    s.z += p[2];
    s.w += p[3];
  }
  const int kl = idx >> 6;  // local k
  float mass = 0.f;
#pragma unroll
  for (int w = 0; w < 8; ++w) mass += ldsM[w * 16 + kl];

  const float* cent = centroid + (size_t)(k0 + kl) * DC + (idx & 63);
  v4f vl;
  vl.x = s.x - mass * cent[0];
  vl.y = s.y - mass * cent[1];
  vl.z = s.z - mass * cent[2];
  vl.w = s.w - mass * cent[3];

  float* o = out + (size_t)b * (KK * DC) + kt * 1024 + idx;
  o[0] = vl.x;
  o[1] = vl.y;
  o[2] = vl.z;
  o[3] = vl.w;

  float ss = vl.x * vl.x + vl.y * vl.y + vl.z * vl.z + vl.w * vl.w;
  atomicAdd(&normAcc[b], ss);
}

// ---------------------------------------------------------------------------
// Kernel 4: L2 normalize each batch row
// ---------------------------------------------------------------------------
__global__ void nv_normalize_kernel(float* __restrict__ out,
                                    const float* __restrict__ normAcc) {
  const int b = blockIdx.x;
  const float nrm = sqrtf(normAcc[b]);
  const float scale = 1.f / fmaxf(nrm, 1e-12f);
  float* o = out + (size_t)b * (KK * DC);
  for (int i = threadIdx.x; i < KK * DC; i += blockDim.x) o[i] *= scale;
}

// ---------------------------------------------------------------------------
extern "C" void kernel_launch(void* const* d_in, const int* in_sizes, int n_in,
                              void* d_out, int out_size, void* d_ws,
                              size_t ws_size, hipStream_t stream) {
  const float* desc = (const float*)d_in[0];
  const float* W1 = (const float*)d_in[1];
  const float* b1 = (const float*)d_in[2];
  const float* W2 = (const float*)d_in[3];
  const float* b2 = (const float*)d_in[4];
  const float* Wa = (const float*)d_in[5];
  const float* ba = (const float*)d_in[6];
  const float* centroid = (const float*)d_in[7];
  float* out = (float*)d_out;

  char* ws = (char*)d_ws;
  float* sq = (float*)ws;                          // 32*4096*64 f32 = 32 MiB
  float* assignT = (float*)(ws + (size_t)33554432);  // 32*128*4096 f32 = 64 MiB
  float* W1T = (float*)(ws + (size_t)100663296);     // 16384 f32
  float* WaT = W1T + DC * DD;                        // 32768 f32
  float* W2T = WaT + KK * DD;                        // 4096 f32
  float* normAcc = W2T + DC * DC;                    // 32 f32

  nv_init_kernel<<<1, 64, 0, stream>>>(normAcc);
  nv_transpose_kernel<<<208, 256, 0, stream>>>(W1, Wa, W2, W1T, WaT, W2T);
  nv_token_kernel<<<(BB * NN) / 128, 256, 0, stream>>>(desc, W1T, b1, W2T, b2,
                                                       WaT, ba, sq, assignT);
  nv_agg_kernel<<<dim3(KK / 16, BB), 256, 0, stream>>>(assignT, sq, centroid,
                                                       out, normAcc);
  nv_normalize_kernel<<<BB, 256, 0, stream>>>(out, normAcc);
}